// LSTMTagger_32873679684062
// MI455X (gfx1250) — compile-verified
//
#include <hip/hip_runtime.h>
#include <hip/hip_bf16.h>
#include <math.h>

typedef __attribute__((ext_vector_type(2))) float v2f;
typedef __attribute__((ext_vector_type(8))) float v8f;

#define INPUT_DIM 22
#define HID 9
#define GATES 36   // 4*HID
#define TAGS 4
#define XGS 48     // padded row stride of x_gates (3 full 16-wide WMMA n-tiles)

// ---------------------------------------------------------------------------
// Kernel 1: x_gates[S][XGS] = x[S][22] @ w_ih.T + (b_ih + b_hh)   (cols 36..47 = 0)
// B (w_ih.T, zero-padded to 24x48) staged in LDS; grid-stride persistent waves
// keep the 18 B fragments in VGPRs; V_WMMA_F32_16X16X4_F32 x18 per 16-row tile.
// Fast path (full tile) has zero lane-dependent branches.
// ---------------------------------------------------------------------------
__global__ __launch_bounds__(256) void xgates_wmma_kernel(
    const float* __restrict__ x, const float* __restrict__ w_ih,
    const float* __restrict__ b_ih, const float* __restrict__ b_hh,
    float* __restrict__ xg, int S)
{
  __shared__ float wpad[48 * 24];   // wpad[n*24+k] = w_ih[n][k], zero padded
  __shared__ float biaspad[48];

  for (int i = threadIdx.x; i < 48 * 24; i += 256) {
    const int n = i / 24, k = i % 24;
    wpad[i] = (n < GATES && k < INPUT_DIM) ? w_ih[n * INPUT_DIM + k] : 0.f;
  }
  if (threadIdx.x < 48) {
    const int n = threadIdx.x;
    biaspad[n] = (n < GATES) ? (b_ih[n] + b_hh[n]) : 0.f;
  }
  __syncthreads();

  const int lane = threadIdx.x & 31;
  const int half = lane >> 4;          // 0: lanes 0-15, 1: lanes 16-31
  const int l16  = lane & 15;

  // B fragments (register-resident across all tiles): B[k][n] = w_ih[n][k]
  // lane<16 -> K=4kk+{0,1}, lane>=16 -> K=4kk+{2,3}; N = 16*nt + l16
  v2f  bfrag[3][6];
  float bias[3];
#pragma unroll
  for (int nt = 0; nt < 3; ++nt) {
    const int n = 16 * nt + l16;
    bias[nt] = biaspad[n];
#pragma unroll
    for (int kk = 0; kk < 6; ++kk)
      bfrag[nt][kk] = *(const v2f*)&wpad[n * 24 + 4 * kk + 2 * half];
  }

  const long nTiles = ((long)S + 15) / 16;
  const long nWaves = (long)gridDim.x * 8;
  for (long tile = (long)blockIdx.x * 8 + (threadIdx.x >> 5);
       tile < nTiles; tile += nWaves) {
    const long row0 = tile * 16;

    if (row0 + 16 <= (long)S) {
      // ---------------- fast path: no lane-dependent control flow ----------
      const float* xrow = x + (row0 + l16) * INPUT_DIM;
      v2f a[6];
#pragma unroll
      for (int kk = 0; kk < 5; ++kk)               // K max = 4*4+2+1 = 19 < 22
        a[kk] = *(const v2f*)(xrow + 4 * kk + 2 * half);
      {
        const v2f t = *(const v2f*)(xrow + 20);    // K = 20,21
        a[5].x = half ? 0.f : t.x;                 // lanes>=16 -> K=22,23 pad
        a[5].y = half ? 0.f : t.y;
      }

#pragma unroll
      for (int nt = 0; nt < 3; ++nt) {
        v8f acc;
#pragma unroll
        for (int v = 0; v < 8; ++v) acc[v] = bias[nt];
#pragma unroll
        for (int kk = 0; kk < 6; ++kk)
          acc = __builtin_amdgcn_wmma_f32_16x16x4_f32(
              false, a[kk], false, bfrag[nt][kk], (short)0, acc, false, false);

        // D: lanes 0-15 -> M=v, lanes 16-31 -> M=v+8; col = 16*nt + l16
        float* xp = xg + (row0 + 8 * half) * XGS + nt * 16 + l16;
#pragma unroll
        for (int v = 0; v < 8; ++v) xp[(long)v * XGS] = acc[v];
      }
    } else {
      // ---------------- guarded remainder tile (rare) ----------------------
      v2f a[6];
#pragma unroll
      for (int kk = 0; kk < 6; ++kk) {
        const long row = row0 + l16;
        const int  k0  = 4 * kk + 2 * half;
        float ax = 0.f, ay = 0.f;
        if (row < S) {
          if (k0 + 0 < INPUT_DIM) ax = x[row * INPUT_DIM + k0 + 0];
          if (k0 + 1 < INPUT_DIM) ay = x[row * INPUT_DIM + k0 + 1];
        }
        a[kk].x = ax; a[kk].y = ay;
      }
#pragma unroll
      for (int nt = 0; nt < 3; ++nt) {
        v8f acc;
#pragma unroll
        for (int v = 0; v < 8; ++v) acc[v] = bias[nt];
#pragma unroll
        for (int kk = 0; kk < 6; ++kk)
          acc = __builtin_amdgcn_wmma_f32_16x16x4_f32(
              false, a[kk], false, bfrag[nt][kk], (short)0, acc, false, false);
        const int n = 16 * nt + l16;
#pragma unroll
        for (int v = 0; v < 8; ++v) {
          const long row = row0 + v + 8 * half;
          if (row < S) xg[row * XGS + n] = acc[v];
        }
      }
    }
  }
}

// ---------------------------------------------------------------------------
// Kernel 2: serial LSTM scan. Single wave32; lane j<9 owns hidden unit j and
// computes its 4 gate pre-activations (w_hh register-resident). x_gates loads
// are software-pipelined one step ahead + hardware prefetch 16 steps ahead.
// ---------------------------------------------------------------------------
__global__ __launch_bounds__(32) void lstm_scan_kernel(
    const float* __restrict__ xg, const float* __restrict__ w_hh,
    float* __restrict__ hs, int S)
{
  const int j  = threadIdx.x;
  const int jj = (j < HID) ? j : 0;   // clamp addresses; lanes>=9 results unused

  float whh[4][HID];                  // whh[k][m] = w_hh[k*HID + j][m]
#pragma unroll
  for (int k = 0; k < 4; ++k)
#pragma unroll
    for (int m = 0; m < HID; ++m)
      whh[k][m] = w_hh[(k * HID + jj) * HID + m];

  float h = 0.f, c = 0.f;

  float xg0 = xg[0 * HID + jj];
  float xg1 = xg[1 * HID + jj];
  float xg2 = xg[2 * HID + jj];
  float xg3 = xg[3 * HID + jj];

  for (int t = 0; t < S; ++t) {
    // deep prefetch (global_prefetch_b8; no LOADcnt cost)
    const long tp = (t + 16 < S) ? (long)(t + 16) : (long)(S - 1);
    __builtin_prefetch(xg + tp * XGS + jj, 0, 0);

    // next timestep's gate inputs (independent of recurrence)
    const long tn = (t + 1 < S) ? (long)(t + 1) : (long)t;
    const float nx0 = xg[tn * XGS + 0 * HID + jj];
    const float nx1 = xg[tn * XGS + 1 * HID + jj];
    const float nx2 = xg[tn * XGS + 2 * HID + jj];
    const float nx3 = xg[tn * XGS + 3 * HID + jj];

    float hm[HID];
#pragma unroll
    for (int m = 0; m < HID; ++m) hm[m] = __shfl(h, m, 32);

    float g0 = xg0, g1 = xg1, g2 = xg2, g3 = xg3;
#pragma unroll
    for (int m = 0; m < HID; ++m) {
      g0 = fmaf(whh[0][m], hm[m], g0);
      g1 = fmaf(whh[1][m], hm[m], g1);
      g2 = fmaf(whh[2][m], hm[m], g2);
      g3 = fmaf(whh[3][m], hm[m], g3);
    }
    const float ig = 1.f / (1.f + __expf(-g0));
    const float fg = 1.f / (1.f + __expf(-g1));
    const float gg = tanhf(g2);
    const float og = 1.f / (1.f + __expf(-g3));
    c = fg * c + ig * gg;
    h = og * tanhf(c);

    if (j < HID) hs[(long)t * HID + j] = h;

    xg0 = nx0; xg1 = nx1; xg2 = nx2; xg3 = nx3;
  }
}

// ---------------------------------------------------------------------------
// Kernel 3: tag = hs @ w_out.T + b_out ; log_softmax over 4 tags.
// One thread per timestep (memory-bound: ~27 MB total).
// ---------------------------------------------------------------------------
__global__ __launch_bounds__(256) void proj_logsoftmax_kernel(
    const float* __restrict__ hs, const float* __restrict__ w_out,
    const float* __restrict__ b_out, float* __restrict__ out, int S)
{
  const long t = (long)blockIdx.x * blockDim.x + threadIdx.x;
  if (t >= S) return;

  float hv[HID];
#pragma unroll
  for (int m = 0; m < HID; ++m) hv[m] = hs[t * HID + m];

  float tag[TAGS];
#pragma unroll
  for (int k = 0; k < TAGS; ++k) {
    float acc = b_out[k];
#pragma unroll
    for (int m = 0; m < HID; ++m) acc = fmaf(w_out[k * HID + m], hv[m], acc);
    tag[k] = acc;
  }

  float mx = tag[0];
#pragma unroll
  for (int k = 1; k < TAGS; ++k) mx = fmaxf(mx, tag[k]);
  float s = 0.f;
#pragma unroll
  for (int k = 0; k < TAGS; ++k) s += __expf(tag[k] - mx);
  const float lse = mx + __logf(s);
#pragma unroll
  for (int k = 0; k < TAGS; ++k) out[t * TAGS + k] = tag[k] - lse;
}

// ---------------------------------------------------------------------------
extern "C" void kernel_launch(void* const* d_in, const int* in_sizes, int n_in,
                              void* d_out, int out_size, void* d_ws, size_t ws_size,
                              hipStream_t stream)
{
  const float* x     = (const float*)d_in[0];
  const float* w_ih  = (const float*)d_in[1];
  const float* w_hh  = (const float*)d_in[2];
  const float* b_ih  = (const float*)d_in[3];
  const float* b_hh  = (const float*)d_in[4];
  const float* w_out = (const float*)d_in[5];
  const float* b_out = (const float*)d_in[6];
  const int S = in_sizes[0] / INPUT_DIM;

  float* xg = (float*)d_ws;                     // [S][XGS], padded cols are 0
  float* hs = xg + (size_t)S * XGS;             // [S][HID]

  const long nTiles = ((long)S + 15) / 16;
  int blocks1 = (int)((nTiles + 7) / 8);
  if (blocks1 > 4096) blocks1 = 4096;           // persistent grid-stride waves
  xgates_wmma_kernel<<<blocks1, 256, 0, stream>>>(x, w_ih, b_ih, b_hh, xg, S);
  lstm_scan_kernel<<<1, 32, 0, stream>>>(xg, w_hh, hs, S);
  proj_logsoftmax_kernel<<<(S + 255) / 256, 256, 0, stream>>>(hs, w_out, b_out,
                                                              (float*)d_out, S);
}